// BilinearMixture_17489106829754
// MI455X (gfx1250) — compile-verified
//
#include <hip/hip_runtime.h>

typedef __attribute__((ext_vector_type(2))) float v2f;
typedef __attribute__((ext_vector_type(8))) float v8f;

#define DIM   128
#define NW    3
#define NC    5

// One wave (32 lanes) processes a tile of 16 edges using f32 WMMA:
//   acc[16x16] = P[16x128] (edges x features, elementwise u*v)
//              x B'[128x16] (B'[d,c] = sum_w W[w,d]*scalars[w,c]; cols >=5 zero)
// so acc column n (n<5) is the class-n logit. Exact f32 math throughout.
__global__ __launch_bounds__(256) void bilinear_mix_wmma_kernel(
    const float* __restrict__ u_feats,
    const float* __restrict__ v_feats,
    const int*   __restrict__ u_idx,
    const int*   __restrict__ v_idx,
    const float* __restrict__ W,        // [NW, DIM]
    const float* __restrict__ scalars,  // [NW, NC]
    const float* __restrict__ u_bias,   // [NUM_USERS, NC]
    const float* __restrict__ v_bias,   // [NUM_ITEMS, NC]
    float*       __restrict__ out,      // [E, NC]
    int E)
{
    const int lane = threadIdx.x & 31;
    const int wave = threadIdx.x >> 5;
    const int wavesPerBlock = blockDim.x >> 5;
    const int n    = lane & 15;          // WMMA N index == class c
    const int koff = (lane >> 4) * 2;    // K sub-offset within a 4-chunk: 0 or 2

    // ---- Precompute register-resident B' fragments (amortized over tiles) ----
    // Fragment j: b[j].x = B'[4j+koff, n], b[j].y = B'[4j+koff+1, n]
    v2f b[DIM / 4];
    {
        float s0 = 0.f, s1 = 0.f, s2 = 0.f;
        if (n < NC) {
            s0 = scalars[0 * NC + n];
            s1 = scalars[1 * NC + n];
            s2 = scalars[2 * NC + n];
        }
        #pragma unroll
        for (int j = 0; j < DIM / 4; ++j) {
            const int d0 = 4 * j + koff;
            b[j].x = W[0 * DIM + d0] * s0 + W[1 * DIM + d0] * s1 + W[2 * DIM + d0] * s2;
            b[j].y = W[0 * DIM + d0 + 1] * s0 + W[1 * DIM + d0 + 1] * s1 + W[2 * DIM + d0 + 1] * s2;
        }
    }

    const int numTiles   = (E + 15) >> 4;
    const int tileStride = gridDim.x * wavesPerBlock;

    for (int tile = blockIdx.x * wavesPerBlock + wave; tile < numTiles; tile += tileStride) {
        const int edgeBase = tile << 4;

        // A-matrix gather addresses: lane L and L+16 cover the same edge row
        // (M = lane&15) at adjacent K halves -> 16B-coalesced row reads.
        int eA = edgeBase + n;
        if (eA >= E) eA = E - 1;
        const long rowU = (long)u_idx[eA];
        const long rowV = (long)v_idx[eA];
        const float* pu = u_feats + rowU * DIM + koff;
        const float* pv = v_feats + rowV * DIM + koff;

        v8f acc = {0.f, 0.f, 0.f, 0.f, 0.f, 0.f, 0.f, 0.f};
        #pragma unroll
        for (int j = 0; j < DIM / 4; ++j) {
            const v2f uu = *(const v2f*)(pu + 4 * j);   // global_load_b64
            const v2f vv = *(const v2f*)(pv + 4 * j);   // global_load_b64
            v2f a;
            a.x = uu.x * vv.x;
            a.y = uu.y * vv.y;
            // D = A(16x4) * B(4x16) + C, full f32 precision
            acc = __builtin_amdgcn_wmma_f32_16x16x4_f32(
                /*neg_a=*/false, a, /*neg_b=*/false, b[j],
                /*c_mod=*/(short)0, acc, /*reuse_a=*/false, /*reuse_b=*/false);
        }

        // ---- Epilogue: lane (half h, n<5) holds out[M = h*8 + r, n] in acc[r] ----
        const int mbase = (lane >> 4) * 8;
        if (n < NC) {
            #pragma unroll
            for (int r = 0; r < 8; ++r) {
                const int e = edgeBase + mbase + r;
                if (e < E) {
                    const long ru = (long)u_idx[e];
                    const long rv = (long)v_idx[e];
                    out[(long)e * NC + n] =
                        acc[r] + u_bias[ru * NC + n] + v_bias[rv * NC + n];
                }
            }
        }
    }
}

extern "C" void kernel_launch(void* const* d_in, const int* in_sizes, int n_in,
                              void* d_out, int out_size, void* d_ws, size_t ws_size,
                              hipStream_t stream) {
    const float* u_feats = (const float*)d_in[0];
    const float* v_feats = (const float*)d_in[1];
    const int*   u_idx   = (const int*)d_in[2];
    const int*   v_idx   = (const int*)d_in[3];
    const float* W       = (const float*)d_in[4];
    const float* scalars = (const float*)d_in[5];
    const float* u_bias  = (const float*)d_in[6];
    const float* v_bias  = (const float*)d_in[7];
    float*       out     = (float*)d_out;

    const int E = in_sizes[2];              // number of edges (u_idx length)
    const int numTiles = (E + 15) / 16;
    const int wavesPerBlock = 8;            // 256 threads = 8 wave32
    int blocks = (numTiles + wavesPerBlock - 1) / wavesPerBlock;
    if (blocks > 4096) blocks = 4096;       // persistent waves; B' amortized
    if (blocks < 1) blocks = 1;

    bilinear_mix_wmma_kernel<<<blocks, 256, 0, stream>>>(
        u_feats, v_feats, u_idx, v_idx, W, scalars, u_bias, v_bias, out, E);
}